// FullDAIN_Layer_52012053954915
// MI455X (gfx1250) — compile-verified
//
#include <hip/hip_runtime.h>
#include <math.h>

#define N_    2048
#define D_    128
#define T_    256
#define DT_   (D_*T_)     // 32768
#define ND_   (N_*D_)     // 262144
#define EPS_  1e-8f
#define NCHUNK 16
#define CH_N  (N_/NCHUNK) // 128

typedef float v2f __attribute__((ext_vector_type(2)));
typedef float v8f __attribute__((ext_vector_type(8)));

// ---------------- helpers ----------------
__device__ __forceinline__ float winsor(float xv, float mu, float al, float be) {
    float eb  = expf(be);
    float w   = eb * tanhf((xv - mu) * expf(-be)) + mu;
    float g   = 1.0f / (1.0f + expf(-al));          // sigmoid(alpha)
    return (1.0f - g) * xv + g * w;
}

__device__ __forceinline__ float yeo_johnson(float y, float lam) {
    if (y >= 0.0f) {
        if (fabsf(lam) > EPS_) return (powf(y + 1.0f, lam) - 1.0f) / lam;
        return log1pf(y);
    } else {
        float l2 = lam - 2.0f;                       // 2-lam = -l2
        if (fabsf(l2) > EPS_) return (powf(1.0f - y, -l2) - 1.0f) / l2;
        return -log1pf(-y);
    }
}

// ---------------- K1a: partial batch mean over N-chunks ----------------
__global__ void k_mean_partial(const float* __restrict__ x, float* __restrict__ part) {
    int dt4   = blockIdx.x * blockDim.x + threadIdx.x;   // 0..8191 (float4 index in D*T)
    int chunk = blockIdx.y;
    const float4* xp = (const float4*)x;
    float4 s = make_float4(0.f, 0.f, 0.f, 0.f);
    int n0 = chunk * CH_N;
    for (int n = 0; n < CH_N; ++n) {
        float4 v = xp[(size_t)(n0 + n) * (DT_ / 4) + dt4];
        s.x += v.x; s.y += v.y; s.z += v.z; s.w += v.w;
    }
    ((float4*)part)[(size_t)chunk * (DT_ / 4) + dt4] = s;
}

// ---------------- K1b: finalize mean ----------------
__global__ void k_mean_final(const float* __restrict__ part, float* __restrict__ mean) {
    int dt4 = blockIdx.x * blockDim.x + threadIdx.x;     // 0..8191
    float4 s = make_float4(0.f, 0.f, 0.f, 0.f);
    for (int c = 0; c < NCHUNK; ++c) {
        float4 v = ((const float4*)part)[(size_t)c * (DT_ / 4) + dt4];
        s.x += v.x; s.y += v.y; s.z += v.z; s.w += v.w;
    }
    const float inv = 1.0f / (float)N_;
    s.x *= inv; s.y *= inv; s.z *= inv; s.w *= inv;
    ((float4*)mean)[dt4] = s;
}

// ---------------- K2: per-(n,d) stats of winsorized x (one wave per row) ----------------
__global__ void k_row_stats(const float* __restrict__ x, const float* __restrict__ mean,
                            const float* __restrict__ alpha, const float* __restrict__ beta,
                            float* __restrict__ avg, float* __restrict__ s2out) {
    int lane = threadIdx.x & 31;
    int wave = threadIdx.x >> 5;
    int row  = blockIdx.x * 8 + wave;                    // row = n*D + d
    int d    = row & (D_ - 1);
    const float4* xr = (const float4*)(x + (size_t)row * T_);
    const float4* mr = (const float4*)(mean  + (size_t)d * T_);
    const float4* ar = (const float4*)(alpha + (size_t)d * T_);
    const float4* br = (const float4*)(beta  + (size_t)d * T_);
    float s1 = 0.f, s2 = 0.f;
    #pragma unroll
    for (int i = 0; i < 2; ++i) {
        int f = lane + i * 32;                           // 64 float4 per row
        float4 xv = xr[f], mu = mr[f], al = ar[f], be = br[f];
        float a = winsor(xv.x, mu.x, al.x, be.x);
        float b = winsor(xv.y, mu.y, al.y, be.y);
        float c = winsor(xv.z, mu.z, al.z, be.z);
        float e = winsor(xv.w, mu.w, al.w, be.w);
        s1 += a + b + c + e;
        s2 += a*a + b*b + c*c + e*e;
    }
    #pragma unroll
    for (int off = 16; off > 0; off >>= 1) {
        s1 += __shfl_xor(s1, off, 32);
        s2 += __shfl_xor(s2, off, 32);
    }
    if (lane == 0) {
        avg[row]   = s1 * (1.0f / (float)T_);
        s2out[row] = s2;
    }
}

// ---------------- K3: Out[n,c] = sum_d A[n,d] * W[c,d]  (fp32 WMMA 16x16x4) ----------------
// mode 0: store raw (shift). mode 1: store guarded reciprocal (1/adaptive_std).
__global__ void k_gemm_wmma(const float* __restrict__ A, const float* __restrict__ W,
                            float* __restrict__ Out, int mode) {
    int lane = threadIdx.x & 31;
    int wave = threadIdx.x >> 5;
    int tile = blockIdx.x * 8 + wave;                    // 0..1023
    int tm = tile >> 3;                                  // 0..127 (row tile)
    int tc = tile & 7;                                   // 0..7   (col tile)
    int m   = lane & 15;
    int grp = lane >> 4;
    int kb  = grp * 2;                                   // K sub-offset {0,2}
    const float* arow = A + (size_t)(tm * 16 + m) * D_;  // A: lane holds row m, K pair
    const float* wrow = W + (size_t)(tc * 16 + m) * D_;  // B[d][c] = W[c][d]: lane = column c
    v8f acc = {0.f, 0.f, 0.f, 0.f, 0.f, 0.f, 0.f, 0.f};
    #pragma unroll 4
    for (int k0 = 0; k0 < D_; k0 += 4) {
        v2f a = *(const v2f*)(arow + k0 + kb);
        v2f b = *(const v2f*)(wrow + k0 + kb);
        acc = __builtin_amdgcn_wmma_f32_16x16x4_f32(false, a, false, b,
                                                    (short)0, acc, false, false);
    }
    int orow0 = tm * 16 + grp * 8;                       // D layout: VGPR j -> row grp*8+j
    int ocol  = tc * 16 + m;
    #pragma unroll
    for (int j = 0; j < 8; ++j) {
        float v = acc[j];
        if (mode == 1) {
            v = (v <= EPS_) ? 1.0f : v;
            v = 1.0f / v;
        }
        Out[(size_t)(orow0 + j) * D_ + ocol] = v;
    }
}

// ---------------- K4: std from expanded moments ----------------
__global__ void k_std(const float* __restrict__ avg, const float* __restrict__ s2,
                      const float* __restrict__ shift, float* __restrict__ stdv) {
    int i = blockIdx.x * blockDim.x + threadIdx.x;
    float a = avg[i], sh = shift[i];
    float var = s2[i] * (1.0f / (float)T_) - 2.0f * sh * a + sh * sh;
    stdv[i] = sqrtf(fmaxf(var, 0.0f) + EPS_);
}

// ---------------- K5: final elementwise pass (recompute winsorize) ----------------
__global__ void k_final(const float* __restrict__ x, const float* __restrict__ mean,
                        const float* __restrict__ alpha, const float* __restrict__ beta,
                        const float* __restrict__ lambd, const float* __restrict__ shift,
                        const float* __restrict__ rinv, float* __restrict__ out) {
    size_t q = (size_t)blockIdx.x * blockDim.x + threadIdx.x;  // float4 index
    size_t e = q * 4;
    int nd = (int)(e >> 8);            // n*D + d  (T = 256)
    int pt = (int)(e & (DT_ - 1));     // d*T + t
    float4 xv = ((const float4*)x)[q];
    float4 mu = *(const float4*)(mean  + pt);
    float4 al = *(const float4*)(alpha + pt);
    float4 be = *(const float4*)(beta  + pt);
    float4 la = *(const float4*)(lambd + pt);
    float sh = shift[nd];
    float ri = rinv[nd];
    float4 o;
    o.x = yeo_johnson((winsor(xv.x, mu.x, al.x, be.x) - sh) * ri, la.x);
    o.y = yeo_johnson((winsor(xv.y, mu.y, al.y, be.y) - sh) * ri, la.y);
    o.z = yeo_johnson((winsor(xv.z, mu.z, al.z, be.z) - sh) * ri, la.z);
    o.w = yeo_johnson((winsor(xv.w, mu.w, al.w, be.w) - sh) * ri, la.w);
    ((float4*)out)[q] = o;
}

// ---------------- launch ----------------
extern "C" void kernel_launch(void* const* d_in, const int* in_sizes, int n_in,
                              void* d_out, int out_size, void* d_ws, size_t ws_size,
                              hipStream_t stream) {
    const float* x     = (const float*)d_in[0];
    const float* alpha = (const float*)d_in[1];
    const float* beta  = (const float*)d_in[2];
    const float* lambd = (const float*)d_in[3];
    const float* Wsh   = (const float*)d_in[4];
    const float* Wsc   = (const float*)d_in[5];
    float* out = (float*)d_out;

    float* ws    = (float*)d_ws;
    float* mean  = ws;                            // DT_
    float* part  = mean + DT_;                    // NCHUNK*DT_
    float* avg   = part + (size_t)NCHUNK * DT_;   // ND_
    float* s2    = avg  + ND_;                    // ND_
    float* shift = s2   + ND_;                    // ND_
    float* stdv  = shift + ND_;                   // ND_
    float* rinv  = stdv + ND_;                    // ND_  (total ~7.5 MB)

    dim3 blk(256);
    k_mean_partial<<<dim3(DT_ / 4 / 256, NCHUNK), blk, 0, stream>>>(x, part);
    k_mean_final  <<<dim3(DT_ / 4 / 256),         blk, 0, stream>>>(part, mean);
    k_row_stats   <<<dim3(ND_ / 8),               blk, 0, stream>>>(x, mean, alpha, beta, avg, s2);
    k_gemm_wmma   <<<dim3(128),                   blk, 0, stream>>>(avg, Wsh, shift, 0);
    k_std         <<<dim3(ND_ / 256),             blk, 0, stream>>>(avg, s2, shift, stdv);
    k_gemm_wmma   <<<dim3(128),                   blk, 0, stream>>>(stdv, Wsc, rinv, 1);
    k_final       <<<dim3((size_t)N_ * DT_ / 4 / 256), blk, 0, stream>>>(
                       x, mean, alpha, beta, lambd, shift, rinv, out);
}